// BiMpmMatching_46471546143268
// MI455X (gfx1250) — compile-verified
//
#include <hip/hip_runtime.h>
#include <math.h>

#define B_   8
#define L_   128
#define D_   512
#define P_   16
#define F_   204
#define EPSF 1e-8f
#define NINF (-1e30f)

typedef float v2f __attribute__((ext_vector_type(2)));
typedef float v8f __attribute__((ext_vector_type(8)));

__device__ __forceinline__ v8f wmma_f32_16x16x4(v2f a, v2f b, v8f c) {
    // D = A(16x4, f32) * B(4x16, f32) + C(16x16, f32)
    return __builtin_amdgcn_wmma_f32_16x16x4_f32(false, a, false, b, (short)0, c,
                                                 false, false);
}

// ---------------- prep kernels ----------------

__global__ void k_maskmul(const float* __restrict__ c1, const float* __restrict__ m1,
                          const float* __restrict__ c2, const float* __restrict__ m2,
                          float* __restrict__ c1m, float* __restrict__ c2m) {
    int idx = blockIdx.x * blockDim.x + threadIdx.x;
    if (idx >= B_ * L_ * D_) return;
    int row = idx >> 9;            // b*L + i
    c1m[idx] = c1[idx] * m1[row];
    c2m[idx] = c2[idx] * m2[row];
}

__global__ void k_rownorm(const float* __restrict__ c1m, const float* __restrict__ c2m,
                          float* __restrict__ n1, float* __restrict__ n2) {
    int r = blockIdx.x;            // 0..2047
    int side = r >> 10;
    int row = r & 1023;            // b*L + i
    const float* src = (side ? c2m : c1m) + row * D_;
    int lane = threadIdx.x;
    float s = 0.f;
    for (int d = lane; d < D_; d += 32) { float v = src[d]; s += v * v; }
    for (int m = 16; m >= 1; m >>= 1) s += __shfl_xor(s, m, 32);
    if (lane == 0) (side ? n2 : n1)[row] = sqrtf(s);
}

__global__ void k_wnorm(const float* __restrict__ c1m, const float* __restrict__ c2m,
                        const float* __restrict__ wmp,
                        float* __restrict__ n1p, float* __restrict__ n2p) {
    int r = blockIdx.x;            // 0..32767
    int side = r >> 14;
    int r2 = r & 16383;
    int b = r2 >> 11;
    int p = (r2 >> 7) & 15;
    int i = r2 & 127;
    const float* c = (side ? c2m : c1m) + (b * L_ + i) * D_;
    const float* w = wmp + p * D_;
    int lane = threadIdx.x;
    float s = 0.f;
    for (int d = lane; d < D_; d += 32) { float v = w[d] * c[d]; s += v * v; }
    for (int m = 16; m >= 1; m >>= 1) s += __shfl_xor(s, m, 32);
    if (lane == 0) (side ? n2p : n1p)[(b * P_ + p) * L_ + i] = sqrtf(s);
}

__global__ void k_w2(const float* __restrict__ wmp, float* __restrict__ w2mp) {
    int idx = blockIdx.x * blockDim.x + threadIdx.x;
    if (idx < P_ * D_) { float w = wmp[idx]; w2mp[idx] = w * w; }
}

__global__ void k_scalars(const float* __restrict__ m1, const float* __restrict__ m2,
                          float* __restrict__ cnt1, float* __restrict__ cnt2,
                          int* __restrict__ il1, int* __restrict__ il2) {
    int t = threadIdx.x;
    if (t >= 16) return;
    int b = t & 7, side = t >> 3;
    const float* m = (side ? m2 : m1) + b * L_;
    float s = 0.f;
    for (int i = 0; i < L_; i++) s += m[i];
    float c = fmaxf(s, EPSF);
    int il = (int)(s + 0.5f) - 1;
    if (il < 0) il = 0;
    if (side) { cnt2[b] = c; il2[b] = il; }
    else      { cnt1[b] = c; il1[b] = il; }
}

// ---------------- cosine = c1m . c2m^T (WMMA f32) ----------------

__global__ void k_cosine(const float* __restrict__ c1m, const float* __restrict__ c2m,
                         const float* __restrict__ n1, const float* __restrict__ n2,
                         float* __restrict__ cosine) {
    int bid = blockIdx.x;          // 512 = b(8) * it(8) * jt(8)
    int b = bid >> 6;
    int it = (bid >> 3) & 7;
    int jt = bid & 7;
    int lane = threadIdx.x;
    int hf = lane >> 4, ln = lane & 15;
    int koff = hf * 2;
    const float* A  = c1m + (b * L_ + it * 16 + ln) * D_;
    const float* Bm = c2m + (b * L_ + jt * 16 + ln) * D_;
    v8f acc = {0.f, 0.f, 0.f, 0.f, 0.f, 0.f, 0.f, 0.f};
    for (int k0 = 0; k0 < D_; k0 += 4) {
        v2f a  = *(const v2f*)(A + k0 + koff);
        v2f bb = *(const v2f*)(Bm + k0 + koff);
        acc = wmma_f32_16x16x4(a, bb, acc);
    }
    int j = jt * 16 + ln;
    float nj = n2[b * L_ + j];
    for (int v = 0; v < 8; v++) {
        int i = it * 16 + v + 8 * hf;
        float ni = n1[b * L_ + i];
        cosine[(b * L_ + i) * L_ + j] = acc[v] / fmaxf(ni * nj, EPSF);
    }
}

// ---------------- cos_max / cos_mean ----------------

__global__ void k_cosred(const float* __restrict__ cosine,
                         const float* __restrict__ m1, const float* __restrict__ m2,
                         const float* __restrict__ cnt1, const float* __restrict__ cnt2,
                         float* __restrict__ out) {
    int r = blockIdx.x;            // 2048
    int side = r >> 10;
    int b = (r >> 7) & 7;
    int row = r & 127;
    int lane = threadIdx.x;
    float mx = NINF, sm = 0.f;
    if (side == 0) {
        const float* cr = cosine + (b * L_ + row) * L_;
        const float* mk = m2 + b * L_;
        for (int j = lane; j < L_; j += 32)
            if (mk[j] > 0.f) { float v = cr[j]; mx = fmaxf(mx, v); sm += v; }
    } else {
        const float* cc = cosine + b * L_ * L_ + row;
        const float* mk = m1 + b * L_;
        for (int i = lane; i < L_; i += 32)
            if (mk[i] > 0.f) { float v = cc[i * L_]; mx = fmaxf(mx, v); sm += v; }
    }
    for (int m = 16; m >= 1; m >>= 1) {
        mx = fmaxf(mx, __shfl_xor(mx, m, 32));
        sm += __shfl_xor(sm, m, 32);
    }
    if (lane == 0) {
        float cnt = side ? cnt1[b] : cnt2[b];
        float* o = out + (b * L_ + row) * F_ + (side ? 102 : 0);
        o[0] = mx;
        o[1] = sm / cnt;
    }
}

// ---------------- s2 = cosine . c2m ; s1 = cosine^T . c1m (WMMA) ----------------

__global__ void k_attgemm(const float* __restrict__ cosine,
                          const float* __restrict__ c1m, const float* __restrict__ c2m,
                          float* __restrict__ s2buf, float* __restrict__ s1buf) {
    int bid = blockIdx.x;          // 4096 = b(8) * side(2) * it(8) * dt(32)
    int dt = bid & 31;
    int it = (bid >> 5) & 7;
    int side = (bid >> 8) & 1;
    int b = bid >> 9;
    int lane = threadIdx.x;
    int hf = lane >> 4, ln = lane & 15;
    int koff = hf * 2;
    const float* C = cosine + b * L_ * L_;
    const float* aBase = side ? (C + it * 16 + ln) : (C + (it * 16 + ln) * L_);
    int aStride = side ? L_ : 1;
    const float* Bm = (side ? c1m : c2m) + b * L_ * D_ + dt * 16 + ln;
    v8f acc = {0.f, 0.f, 0.f, 0.f, 0.f, 0.f, 0.f, 0.f};
    for (int k0 = 0; k0 < L_; k0 += 4) {
        int k = k0 + koff;
        v2f a, bb;
        a.x = aBase[k * aStride];
        a.y = aBase[(k + 1) * aStride];
        bb.x = Bm[k * D_];
        bb.y = Bm[(k + 1) * D_];
        acc = wmma_f32_16x16x4(a, bb, acc);
    }
    float* S = (side ? s1buf : s2buf) + b * L_ * D_;
    for (int v = 0; v < 8; v++) {
        int rr = it * 16 + v + 8 * hf;
        S[rr * D_ + dt * 16 + ln] = acc[v];
    }
}

// ---------------- softmax over D (in place) ----------------

__global__ void k_softmax(const float* __restrict__ m1, const float* __restrict__ m2,
                          float* __restrict__ s2buf, float* __restrict__ s1buf) {
    int r = blockIdx.x;            // 2048
    int side = r >> 10;
    int b = (r >> 7) & 7;
    int row = r & 127;
    float* buf = (side ? s1buf : s2buf) + (b * L_ + row) * D_;
    const float* mk = side ? m2 : m1;
    int lane = threadIdx.x;
    if (mk[b * L_ + row] <= 0.f) {   // uniform (all-INF row -> softmax = 1/D)
        float u = 1.0f / (float)D_;
        for (int d = lane; d < D_; d += 32) buf[d] = u;
        return;
    }
    float x[16];
    float mx = NINF;
    for (int t = 0; t < 16; t++) { x[t] = buf[lane + 32 * t]; mx = fmaxf(mx, x[t]); }
    for (int m = 16; m >= 1; m >>= 1) mx = fmaxf(mx, __shfl_xor(mx, m, 32));
    float sm = 0.f;
    for (int t = 0; t < 16; t++) { x[t] = __expf(x[t] - mx); sm += x[t]; }
    for (int m = 16; m >= 1; m >>= 1) sm += __shfl_xor(sm, m, 32);
    float inv = 1.0f / sm;
    for (int t = 0; t < 16; t++) buf[lane + 32 * t] = x[t] * inv;
}

// ---------------- att_max over the opposite sequence ----------------

__global__ void k_attmax(const float* __restrict__ cosine,
                         const float* __restrict__ c1m, const float* __restrict__ c2m,
                         const float* __restrict__ m1, const float* __restrict__ m2,
                         float* __restrict__ amax2, float* __restrict__ amax1) {
    int idx = blockIdx.x * blockDim.x + threadIdx.x;
    if (idx >= B_ * 2 * L_ * D_) return;
    int d = idx & 511;
    int row = (idx >> 9) & 127;
    int b = (idx >> 16) & 7;
    int side = idx >> 19;
    float mx = NINF;
    if (side == 0) {
        const float* cr = cosine + (b * L_ + row) * L_;
        const float* mk = m2 + b * L_;
        const float* cv = c2m + b * L_ * D_ + d;
        for (int j = 0; j < L_; j++)
            if (mk[j] > 0.f) mx = fmaxf(mx, cv[j * D_] * cr[j]);
        amax2[(b * L_ + row) * D_ + d] = mx;
    } else {
        const float* cc = cosine + b * L_ * L_ + row;
        const float* mk = m1 + b * L_;
        const float* cv = c1m + b * L_ * D_ + d;
        for (int i = 0; i < L_; i++)
            if (mk[i] > 0.f) mx = fmaxf(mx, cv[i * D_] * cc[i * L_]);
        amax1[(b * L_ + row) * D_ + d] = mx;
    }
}

// ---------------- maxpool match: per (b,p) GEMM + fused reductions (WMMA) ----------------

__global__ void k_maxpool(const float* __restrict__ c1m, const float* __restrict__ c2m,
                          const float* __restrict__ w2mp,
                          const float* __restrict__ n1p, const float* __restrict__ n2p,
                          const float* __restrict__ m1, const float* __restrict__ m2,
                          const float* __restrict__ cnt1, const float* __restrict__ cnt2,
                          float* __restrict__ out) {
    __shared__ float colmax[8][L_];
    __shared__ float colsum[8][L_];
    int bid = blockIdx.x;          // 128 = b(8) * p(16)
    int b = bid >> 4, p = bid & 15;
    int tid = threadIdx.x;
    int w = tid >> 5, lane = tid & 31;
    int hf = lane >> 4, ln = lane & 15;
    int koff = hf * 2;
    int i0 = w * 16;
    const float* W = w2mp + p * D_;
    const float* A = c1m + (b * L_ + i0 + ln) * D_;
    float rmax[8], rsum[8], n1v[8], m1v[8];
    for (int v = 0; v < 8; v++) {
        rmax[v] = NINF; rsum[v] = 0.f;
        int i = i0 + v + 8 * hf;
        n1v[v] = n1p[(b * P_ + p) * L_ + i];
        m1v[v] = m1[b * L_ + i];
    }
    for (int jt = 0; jt < 8; jt++) {
        int j0 = jt * 16;
        const float* Bm = c2m + (b * L_ + j0 + ln) * D_;
        v8f acc = {0.f, 0.f, 0.f, 0.f, 0.f, 0.f, 0.f, 0.f};
        for (int k0 = 0; k0 < D_; k0 += 4) {
            v2f a, bb;
            a.x = A[k0 + koff] * W[k0 + koff];
            a.y = A[k0 + koff + 1] * W[k0 + koff + 1];
            bb = *(const v2f*)(Bm + k0 + koff);
            acc = wmma_f32_16x16x4(a, bb, acc);
        }
        int j = j0 + ln;
        float n2v = n2p[(b * P_ + p) * L_ + j];
        float m2v = m2[b * L_ + j];
        float cmx = NINF, csm = 0.f;
        for (int v = 0; v < 8; v++) {
            float mv = acc[v] / fmaxf(n1v[v] * n2v, EPSF);
            if (m2v > 0.f) { rmax[v] = fmaxf(rmax[v], mv); rsum[v] += mv; }
            if (m1v[v] > 0.f) { cmx = fmaxf(cmx, mv); csm += mv; }
        }
        cmx = fmaxf(cmx, __shfl_xor(cmx, 16, 32));
        csm += __shfl_xor(csm, 16, 32);
        if (hf == 0) { colmax[w][j] = cmx; colsum[w][j] = csm; }
    }
    for (int m = 1; m <= 8; m <<= 1) {
        for (int v = 0; v < 8; v++) {
            rmax[v] = fmaxf(rmax[v], __shfl_xor(rmax[v], m, 32));
            rsum[v] += __shfl_xor(rsum[v], m, 32);
        }
    }
    if (ln == 0) {
        float c2f = cnt2[b];
        for (int v = 0; v < 8; v++) {
            int i = i0 + v + 8 * hf;
            float* o = out + (b * L_ + i) * F_;
            o[36 + p] = rmax[v];
            o[52 + p] = rsum[v] / c2f;
        }
    }
    __syncthreads();
    if (tid < L_) {
        int j = tid;
        float mx = NINF, sm = 0.f;
        for (int ww = 0; ww < 8; ww++) { mx = fmaxf(mx, colmax[ww][j]); sm += colsum[ww][j]; }
        float* o = out + (b * L_ + j) * F_;
        o[138 + p] = mx;
        o[154 + p] = sm / cnt1[b];
    }
}

// ---------------- per-position multi-perspective match ----------------

__global__ void k_match(const float* __restrict__ c1m, const float* __restrict__ c2m,
                        const float* __restrict__ s2buf, const float* __restrict__ s1buf,
                        const float* __restrict__ amax2, const float* __restrict__ amax1,
                        const float* __restrict__ wf, const float* __restrict__ wb,
                        const float* __restrict__ wa, const float* __restrict__ wx,
                        const int* __restrict__ il1, const int* __restrict__ il2,
                        float* __restrict__ out) {
    __shared__ float su[D_];
    __shared__ float sv[4][D_];
    int bid = blockIdx.x;          // 2048
    int side = bid >> 10;
    int b = (bid >> 7) & 7;
    int row = bid & 127;
    int tid = threadIdx.x;
    const float *u, *va, *vb, *vc, *vd;
    if (side == 0) {
        u  = c1m + (b * L_ + row) * D_;
        va = c2m + (b * L_ + il2[b]) * D_;   // c2_last
        vb = c2m + (b * L_) * D_;            // c2_first
        vc = s2buf + (b * L_ + row) * D_;    // att_mean_2
        vd = amax2 + (b * L_ + row) * D_;    // att_max_2
    } else {
        u  = c2m + (b * L_ + row) * D_;
        va = c1m + (b * L_ + il1[b]) * D_;   // c1_last
        vb = c1m + (b * L_) * D_;            // c1_first
        vc = s1buf + (b * L_ + row) * D_;    // att_mean_1
        vd = amax1 + (b * L_ + row) * D_;    // att_max_1
    }
    for (int d = tid; d < D_; d += 128) {
        su[d] = u[d]; sv[0][d] = va[d]; sv[1][d] = vb[d]; sv[2][d] = vc[d]; sv[3][d] = vd[d];
    }
    __syncthreads();
    int w = tid >> 5, lane = tid & 31;
    int p = lane & 15, h = lane >> 4;
    const float* wsel = (w == 0) ? wf : (w == 1) ? wb : (w == 2) ? wa : wx;
    const float* wp = wsel + p * D_;
    const float* V = sv[w];
    float s_uvw = 0.f, s_uuw = 0.f, s_vvw = 0.f, s_uv = 0.f, s_uu = 0.f, s_vv = 0.f;
    for (int t = 0; t < 256; t++) {
        int d = h * 256 + t;
        float uu = su[d], vv = V[d];
        float ww = wp[d];
        float w2 = ww * ww;
        s_uvw += w2 * uu * vv; s_uuw += w2 * uu * uu; s_vvw += w2 * vv * vv;
        s_uv  += uu * vv;      s_uu  += uu * uu;      s_vv  += vv * vv;
    }
    s_uvw += __shfl_xor(s_uvw, 16, 32);
    s_uuw += __shfl_xor(s_uuw, 16, 32);
    s_vvw += __shfl_xor(s_vvw, 16, 32);
    s_uv  += __shfl_xor(s_uv, 16, 32);
    s_uu  += __shfl_xor(s_uu, 16, 32);
    s_vv  += __shfl_xor(s_vv, 16, 32);
    int base = ((w == 0) ? 2 : (w == 1) ? 19 : (w == 2) ? 68 : 85) + (side ? 102 : 0);
    float* o = out + (b * L_ + row) * F_;
    if (lane == 0)
        o[base] = s_uv / fmaxf(sqrtf(s_uu) * sqrtf(s_vv), EPSF);
    if (h == 0)
        o[base + 1 + p] = s_uvw / fmaxf(sqrtf(s_uuw) * sqrtf(s_vvw), EPSF);
}

// ---------------- launch ----------------

extern "C" void kernel_launch(void* const* d_in, const int* in_sizes, int n_in,
                              void* d_out, int out_size, void* d_ws, size_t ws_size,
                              hipStream_t stream) {
    const float* ctx1 = (const float*)d_in[0];
    const float* m1   = (const float*)d_in[1];
    const float* ctx2 = (const float*)d_in[2];
    const float* m2   = (const float*)d_in[3];
    const float* wf   = (const float*)d_in[4];
    const float* wb   = (const float*)d_in[5];
    const float* wmp  = (const float*)d_in[6];
    const float* wa   = (const float*)d_in[7];
    const float* wx   = (const float*)d_in[8];
    float* out = (float*)d_out;

    float* ws = (float*)d_ws;
    const size_t N_CD = (size_t)B_ * L_ * D_;   // 524288
    float* c1m    = ws;
    float* c2m    = c1m + N_CD;
    float* cosine = c2m + N_CD;                 // B*L*L = 131072
    float* s2buf  = cosine + (size_t)B_ * L_ * L_;
    float* s1buf  = s2buf + N_CD;
    float* amax2  = s1buf + N_CD;
    float* amax1  = amax2 + N_CD;
    float* n1     = amax1 + N_CD;               // B*L
    float* n2     = n1 + B_ * L_;
    float* n1p    = n2 + B_ * L_;               // B*P*L = 16384
    float* n2p    = n1p + B_ * P_ * L_;
    float* w2mp   = n2p + B_ * P_ * L_;         // P*D = 8192
    float* cnt1   = w2mp + P_ * D_;
    float* cnt2   = cnt1 + B_;
    int*   il1    = (int*)(cnt2 + B_);
    int*   il2    = il1 + B_;

    k_maskmul<<<(B_ * L_ * D_ + 255) / 256, 256, 0, stream>>>(ctx1, m1, ctx2, m2, c1m, c2m);
    k_rownorm<<<2048, 32, 0, stream>>>(c1m, c2m, n1, n2);
    k_wnorm<<<32768, 32, 0, stream>>>(c1m, c2m, wmp, n1p, n2p);
    k_w2<<<(P_ * D_ + 255) / 256, 256, 0, stream>>>(wmp, w2mp);
    k_scalars<<<1, 16, 0, stream>>>(m1, m2, cnt1, cnt2, il1, il2);
    k_cosine<<<512, 32, 0, stream>>>(c1m, c2m, n1, n2, cosine);
    k_cosred<<<2048, 32, 0, stream>>>(cosine, m1, m2, cnt1, cnt2, out);
    k_attgemm<<<4096, 32, 0, stream>>>(cosine, c1m, c2m, s2buf, s1buf);
    k_softmax<<<2048, 32, 0, stream>>>(m1, m2, s2buf, s1buf);
    k_attmax<<<(B_ * 2 * L_ * D_ + 255) / 256, 256, 0, stream>>>(cosine, c1m, c2m, m1, m2, amax2, amax1);
    k_maxpool<<<B_ * P_, 256, 0, stream>>>(c1m, c2m, w2mp, n1p, n2p, m1, m2, cnt1, cnt2, out);
    k_match<<<2048, 128, 0, stream>>>(c1m, c2m, s2buf, s1buf, amax2, amax1,
                                      wf, wb, wa, wx, il1, il2, out);
}